// Selection_62878321214039
// MI455X (gfx1250) — compile-verified
//
#include <hip/hip_runtime.h>
#include <hip/hip_bf16.h>

typedef __attribute__((ext_vector_type(2))) float v2f;
typedef __attribute__((ext_vector_type(8))) float v8f;

#define N_PTS   56564
#define BATCH   16
#define KSEL    8
#define CHAN    128
#define NSEG    128
// per-segment n count, multiple of 16 (main loop step)
#define NPER    448

__global__ void zero_out_kernel(float* __restrict__ out, int n) {
    int i = blockIdx.x * blockDim.x + threadIdx.x;
    if (i < n) out[i] = 0.0f;
}

// idx[n] = argmax_k pw[k*N + n]  (first max wins, matching jnp.argmax)
__global__ void argmax_kernel(const float* __restrict__ pw,
                              int* __restrict__ idx, int N) {
    int n = blockIdx.x * blockDim.x + threadIdx.x;
    if (n >= N) return;
    float best = pw[n];
    int bk = 0;
#pragma unroll
    for (int k = 1; k < KSEL; ++k) {
        float v = pw[k * N + n];
        if (v > best) { best = v; bk = k; }
    }
    idx[n] = bk;
}

// One 16x16x4 step: A = one-hot rows from 4 argmax indices, B = 4x16 x-slice.
__device__ __forceinline__ v8f wmma_step(const float* __restrict__ xb, int4 e,
                                         int n0, int half, int m, v8f acc) {
    // B 4x16: VGPR0 = rows {0,2}, VGPR1 = rows {1,3}; row r -> n0+r
    int nb = n0 + 2 * half;
    float bx0 = __builtin_nontemporal_load(xb + (size_t)nb * CHAN);
    float bx1 = __builtin_nontemporal_load(xb + (size_t)(nb + 1) * CHAN);
    // A 16x4 one-hot: VGPR0 = K {0,2}, VGPR1 = K {1,3}, row = m
    int e_lo = half ? e.z : e.x;
    int e_hi = half ? e.w : e.y;
    v2f a;
    a.x = (e_lo == m) ? 1.0f : 0.0f;
    a.y = (e_hi == m) ? 1.0f : 0.0f;
    v2f bb;
    bb.x = bx0;
    bb.y = bx1;
    return __builtin_amdgcn_wmma_f32_16x16x4_f32(
        false, a, false, bb, (short)0, acc, false, false);
}

// out[b,k,c] += sum over this block's n-segment of onehot(idx[n]==k) * x[b,n,c]
__global__ void __launch_bounds__(256)
scatter_wmma_kernel(const float* __restrict__ x,
                    const int*   __restrict__ idx,
                    float*       __restrict__ out) {
    const int b     = blockIdx.y;
    const int lane  = threadIdx.x & 31;
    const int wave  = threadIdx.x >> 5;   // 8 waves -> 8 c-tiles of 16
    const int half  = lane >> 4;          // K-half selector (ISA 32-bit A/B layout)
    const int m     = lane & 15;          // A row / D column within tile
    const int ctile = wave * 16;

    int n_begin = blockIdx.x * NPER;
    int n_end   = n_begin + NPER;
    if (n_end > N_PTS) n_end = N_PTS;     // uniform per block; EXEC stays full

    // lane-private column pointer into x[b, :, ctile+m]
    const float* xb = x + (size_t)b * N_PTS * CHAN + (ctile + m);

    v8f acc0 = {};
    v8f acc1 = {};

    int n0 = n_begin;
    // Main loop: 16 n per iteration = 4 WMMA steps, all loads issued up front,
    // two independent accumulator chains for pipelining.
    for (; n0 + 16 <= n_end; n0 += 16) {
        int4 e0 = *(const int4*)(idx + n0);       // contiguous 64B, uniform ->
        int4 e1 = *(const int4*)(idx + n0 + 4);   // merges into wide s_load
        int4 e2 = *(const int4*)(idx + n0 + 8);
        int4 e3 = *(const int4*)(idx + n0 + 12);
        acc0 = wmma_step(xb, e0, n0,      half, m, acc0);
        acc1 = wmma_step(xb, e1, n0 + 4,  half, m, acc1);
        acc0 = wmma_step(xb, e2, n0 + 8,  half, m, acc0);
        acc1 = wmma_step(xb, e3, n0 + 12, half, m, acc1);
    }
    // Uniform tail (segment sizes are multiples of 4)
    for (; n0 < n_end; n0 += 4) {
        int4 e = *(const int4*)(idx + n0);
        acc0 = wmma_step(xb, e, n0, half, m, acc0);
    }

    v8f acc = acc0 + acc1;

    // D 16x16: VGPR v holds row v (lanes 0-15) and row v+8 (lanes 16-31).
    // Real outputs are rows k=0..7 -> lanes 0..15, col = ctile + lane.
    if (lane < 16) {
        float* ob = out + ((size_t)b * KSEL) * CHAN + ctile + m;
#pragma unroll
        for (int k = 0; k < KSEL; ++k) {
            atomicAdd(ob + k * CHAN, acc[k]);
        }
    }
}

extern "C" void kernel_launch(void* const* d_in, const int* in_sizes, int n_in,
                              void* d_out, int out_size, void* d_ws, size_t ws_size,
                              hipStream_t stream) {
    const float* x  = (const float*)d_in[0];   // (B, N, C) f32
    const float* pw = (const float*)d_in[1];   // (1, K, N, 1) f32 -> [k*N + n]
    float* out = (float*)d_out;                // (B, K, C) f32
    int* idx = (int*)d_ws;                     // N ints of scratch

    const int out_elems = BATCH * KSEL * CHAN; // 16384

    zero_out_kernel<<<(out_elems + 255) / 256, 256, 0, stream>>>(out, out_elems);
    argmax_kernel<<<(N_PTS + 255) / 256, 256, 0, stream>>>(pw, idx, N_PTS);

    dim3 grid(NSEG, BATCH);
    scatter_wmma_kernel<<<grid, 256, 0, stream>>>(x, idx, out);
}